// RewardLoss_22265110462728
// MI455X (gfx1250) — compile-verified
//
#include <hip/hip_runtime.h>

#define TPB 256
#define NWAVES (TPB / 32)

typedef float v2f __attribute__((ext_vector_type(2)));
typedef float v4f __attribute__((ext_vector_type(4)));
typedef float v8f __attribute__((ext_vector_type(8)));

// ---------- top-5 (value, index) kept sorted descending; tie -> lower index ----------
struct Top5 { float v[5]; unsigned i[5]; };

__device__ __forceinline__ bool t5_better(float v1, unsigned i1, float v2, unsigned i2) {
  return (v1 > v2) || ((v1 == v2) && (i1 < i2));
}

__device__ __forceinline__ void t5_init(Top5& t) {
#pragma unroll
  for (int j = 0; j < 5; ++j) { t.v[j] = -__builtin_inff(); t.i[j] = 0xFFFFFFFFu; }
}

__device__ __forceinline__ void t5_insert(Top5& t, float v, unsigned idx) {
  if (!t5_better(v, idx, t.v[4], t.i[4])) return;   // common case: one cmp + skip
  t.v[4] = v; t.i[4] = idx;
#pragma unroll
  for (int j = 4; j > 0; --j) {
    if (t5_better(t.v[j], t.i[j], t.v[j - 1], t.i[j - 1])) {
      float tv = t.v[j]; t.v[j] = t.v[j - 1]; t.v[j - 1] = tv;
      unsigned ti = t.i[j]; t.i[j] = t.i[j - 1]; t.i[j - 1] = ti;
    }
  }
}

// XOR-butterfly merge across the wave32. Partner subcubes are disjoint lane sets,
// so real candidate lists never merge with overlapping copies; sentinel entries
// (-inf, ~0u) never beat nor duplicate past real entries. All lanes converge to
// the identical wave-wide top-5.
__device__ __forceinline__ void t5_wave_merge(Top5& t) {
#pragma unroll
  for (int off = 1; off < 32; off <<= 1) {
    Top5 o;
#pragma unroll
    for (int j = 0; j < 5; ++j) {
      o.v[j] = __shfl_xor(t.v[j], off, 32);
      o.i[j] = (unsigned)__shfl_xor((int)t.i[j], off, 32);
    }
#pragma unroll
    for (int j = 0; j < 5; ++j) t5_insert(t, o.v[j], o.i[j]);
  }
}

// ---------- wave32 sum via V_WMMA_F32_16X16X4_F32 (full-precision f32, RNE) ----------
// A (16x4 f32): lanes 0-15 hold {K=0,K=1}, lanes 16-31 hold {K=2,K=3}.
// s in first slot, 0 in second; B = all-ones (layout agnostic).
// D[m][n] = s_m + s_{m+16}; sum the 8 acc VGPRs + shfl_xor(16) -> full wave sum.
__device__ __forceinline__ float wave_sum32(float s) {
  v2f a; a[0] = s;    a[1] = 0.0f;
  v2f b; b[0] = 1.0f; b[1] = 1.0f;
  v8f c = {};
  v8f d = __builtin_amdgcn_wmma_f32_16x16x4_f32(false, a, false, b, (short)0, c, false, false);
  float t = ((d[0] + d[1]) + (d[2] + d[3])) + ((d[4] + d[5]) + (d[6] + d[7]));
  t += __shfl_xor(t, 16, 32);
  return t;
}

// ---------- adjustment terms (match reference constants exactly) ----------
__device__ __forceinline__ float adj_rest(float r) {
  return (r >= 2.0f) ? (0.5f * 0.3f * fabsf(r) / 100.0f) : 0.0f;
}
__device__ __forceinline__ float adj_strict(float r) {
  float ar = fabsf(r);
  if (r < 0.0f)  return 0.5f * ar / 100.0f;
  if (r >= 2.0f) return -(1.0f * 2.0f) * r / 100.0f;
  if (r > 0.0f)  return -1.0f * r / 100.0f;
  return 0.0f;
}
__device__ __forceinline__ float adj_mid(float r) {
  float ar = fabsf(r);
  if (r < 0.0f)  return 0.5f * 0.5f * ar / 100.0f;
  if (r >= 2.0f) return -1.0f * r / 100.0f;
  if (r > 0.0f)  return -0.5f * r / 100.0f;
  return 0.0f;
}

// ---------- block-level combine: every wave redundantly reduces the NWAVES
// per-wave results (lanes 0..NWAVES-1 carry them, others carry sentinels), so
// all WMMAs execute at top level with EXEC all-ones. No serialized scalar merge.
__device__ __forceinline__ void block_combine(
    float wsum, const Top5& t,
    float* s_sum, float (*s_val)[5], unsigned (*s_idx)[5],
    float& block_sum, Top5& block_top) {
  const int wave = threadIdx.x >> 5;
  const int lane = threadIdx.x & 31;
  if (lane == 0) {
    s_sum[wave] = wsum;
#pragma unroll
    for (int j = 0; j < 5; ++j) { s_val[wave][j] = t.v[j]; s_idx[wave][j] = t.i[j]; }
  }
  __syncthreads();
  const int  w    = lane & (NWAVES - 1);
  const bool live = lane < NWAVES;
  float cs = live ? s_sum[w] : 0.0f;
  block_sum = wave_sum32(cs);            // EXEC all-ones; every wave computes it
  Top5 m;
#pragma unroll
  for (int j = 0; j < 5; ++j) {
    m.v[j] = live ? s_val[w][j] : -__builtin_inff();
    m.i[j] = live ? s_idx[w][j] : 0xFFFFFFFFu;
  }
  t5_wave_merge(m);
  block_top = m;
}

// ---------- pass 1: streaming sum of base+rest_adj, per-block top-5 ----------
__global__ __launch_bounds__(TPB) void rl_pass1(
    const float* __restrict__ pred, const float* __restrict__ tru,
    const float* __restrict__ ret, float* __restrict__ ws_sum,
    float* __restrict__ ws_val, unsigned* __restrict__ ws_idx, int nvec) {
  __shared__ float s_sum[NWAVES];
  __shared__ float s_val[NWAVES][5];
  __shared__ unsigned s_idx[NWAVES][5];

  const int tid = blockIdx.x * TPB + threadIdx.x;
  const int stride = gridDim.x * TPB;

  const v4f* p4 = (const v4f*)pred;
  const v4f* q4 = (const v4f*)tru;
  const v4f* r4 = (const v4f*)ret;

  float s = 0.0f;
  Top5 t; t5_init(t);

  for (int v = tid; v < nvec; v += stride) {
    // speculative prefetch of next tile (silently dropped if past the buffer)
    __builtin_prefetch(&p4[v + stride], 0, 0);   // global_prefetch_b8
    __builtin_prefetch(&q4[v + stride], 0, 0);
    __builtin_prefetch(&r4[v + stride], 0, 0);
    // single-use streams -> non-temporal 128-bit loads (evict-first in L2)
    v4f p = __builtin_nontemporal_load(&p4[v]);
    v4f q = __builtin_nontemporal_load(&q4[v]);
    v4f r = __builtin_nontemporal_load(&r4[v]);
    float d0 = p[0] - q[0], d1 = p[1] - q[1], d2 = p[2] - q[2], d3 = p[3] - q[3];
    s += d0 * d0 + adj_rest(r[0]);
    s += d1 * d1 + adj_rest(r[1]);
    s += d2 * d2 + adj_rest(r[2]);
    s += d3 * d3 + adj_rest(r[3]);
    unsigned base = 4u * (unsigned)v;
    t5_insert(t, p[0], base + 0u);
    t5_insert(t, p[1], base + 1u);
    t5_insert(t, p[2], base + 2u);
    t5_insert(t, p[3], base + 3u);
  }

  float wsum = wave_sum32(s);   // loop reconverged -> EXEC all-ones
  t5_wave_merge(t);

  float block_sum; Top5 bt;
  block_combine(wsum, t, s_sum, s_val, s_idx, block_sum, bt);

  if (threadIdx.x == 0) {
    ws_sum[blockIdx.x] = block_sum;
#pragma unroll
    for (int j = 0; j < 5; ++j) {
      ws_val[blockIdx.x * 5 + j] = bt.v[j];
      ws_idx[blockIdx.x * 5 + j] = bt.i[j];
    }
  }
}

// ---------- pass 2: deterministic final merge + rank correction ----------
__global__ __launch_bounds__(TPB) void rl_pass2(
    const float* __restrict__ ret, const float* __restrict__ ws_sum,
    const float* __restrict__ ws_val, const unsigned* __restrict__ ws_idx,
    float* __restrict__ out, int nblk, int n) {
  __shared__ float s_sum[NWAVES];
  __shared__ float s_val[NWAVES][5];
  __shared__ unsigned s_idx[NWAVES][5];

  float s = 0.0f;
  Top5 t; t5_init(t);
  for (int i = threadIdx.x; i < nblk; i += TPB) {  // nblk multiple of 256 -> uniform
    s += ws_sum[i];
#pragma unroll
    for (int j = 0; j < 5; ++j) t5_insert(t, ws_val[i * 5 + j], ws_idx[i * 5 + j]);
  }
  float wsum = wave_sum32(s);
  t5_wave_merge(t);

  float total; Top5 ft;
  block_combine(wsum, t, s_sum, s_val, s_idx, total, ft);

  if (threadIdx.x == 0) {
    // ranks 1..3 -> strict, 4..5 -> mid; pass 1 charged them rest_adj
    float delta = 0.0f;
#pragma unroll
    for (int rk = 0; rk < 5; ++rk) {
      unsigned idx = ft.i[rk];
      if (idx < (unsigned)n) {
        float r = ret[idx];
        float target = (rk < 3) ? adj_strict(r) : adj_mid(r);
        delta += target - adj_rest(r);
      }
    }
    out[0] = (total + delta) / (float)n;
  }
}

extern "C" void kernel_launch(void* const* d_in, const int* in_sizes, int n_in,
                              void* d_out, int out_size, void* d_ws, size_t ws_size,
                              hipStream_t stream) {
  const float* pred = (const float*)d_in[0];
  const float* tru  = (const float*)d_in[1];
  const float* ret  = (const float*)d_in[2];
  const int n = in_sizes[0];
  const int nvec = n >> 2;  // n = 2^24, float4 granularity

  int nblk = 4096;  // power of two >= 256 keeps pass-2 trips wave-uniform
  while ((size_t)nblk * 44u > ws_size && nblk > 256) nblk >>= 1;

  float*    ws_sum = (float*)d_ws;
  float*    ws_val = ws_sum + nblk;
  unsigned* ws_idx = (unsigned*)(ws_val + 5 * nblk);

  rl_pass1<<<nblk, TPB, 0, stream>>>(pred, tru, ret, ws_sum, ws_val, ws_idx, nvec);
  rl_pass2<<<1, TPB, 0, stream>>>(ret, ws_sum, ws_val, ws_idx, (float*)d_out, nblk, n);
}